// DriftingLoss_47442208752032
// MI455X (gfx1250) — compile-verified
//
#include <hip/hip_runtime.h>
#include <math.h>

#define G   4096
#define GP  8192
#define DD  256
#define ND  (G * DD)   // 1048576

// workspace layout (float offsets)
#define N2_OFF   0          // 8192  : squared norms of targets (first G == gen norms)
#define RS_OFF   8192       // 3*4096: kernel row sums per temp
#define CS_OFF   20480      // 3*8192: kernel col sums per temp
#define NRS_OFF  45056      // 3*4096: neg_row_sum (j <  G) of normalized kernel
#define PRS_OFF  57344      // 3*4096: pos_row_sum (j >= G) of normalized kernel
#define SSQ_OFF  69632      // 3     : sum of squares of V_tau
#define ACC_OFF  69635      // 1     : final accumulator
#define ZERO_BEG RS_OFF
#define ZERO_LEN (ACC_OFF + 1 - RS_OFF)
#define V_OFF    131072     // 3*ND  : V_tau buffers

typedef float v2f __attribute__((ext_vector_type(2)));
typedef float v8f __attribute__((ext_vector_type(8)));

__constant__ float INVTAU[3] = {50.0f, 20.0f, 5.0f};  // 1/0.02, 1/0.05, 1/0.2

// ~1-ulp hardware sqrt/rsq: inputs here are clamped to [0, ~1e3] / [1e-12, inf),
// well inside the range where v_sqrt_f32 / v_rsq_f32 need no IEEE fixup path.
__device__ __forceinline__ float fast_sqrt(float x) {
#if __has_builtin(__builtin_amdgcn_sqrtf)
  return __builtin_amdgcn_sqrtf(x);
#else
  return sqrtf(x);
#endif
}
__device__ __forceinline__ float fast_rsqrt(float x) {
#if __has_builtin(__builtin_amdgcn_rsqf)
  return __builtin_amdgcn_rsqf(x);
#else
  return rsqrtf(x);
#endif
}

// Accumulate C += A(16xK slice) * B(Kx16) with fp32 WMMA 16x16x4, operands in
// global memory. A lane(l): row m=l&15, K={2h,2h+1}+k0 (h=l>>4) -> float2.
__device__ __forceinline__ v8f wmma_stile(const float* __restrict__ gA,
                                          const float* __restrict__ gB,
                                          int kbeg, int kend, v8f c) {
#pragma unroll 8
  for (int k = kbeg; k < kend; k += 4) {
    v2f a = *(const v2f*)(gA + k);
    v2f b = *(const v2f*)(gB + k);
    c = __builtin_amdgcn_wmma_f32_16x16x4_f32(false, a, false, b, (short)0, c,
                                              false, false);
  }
  return c;
}

__global__ __launch_bounds__(256) void zero_kernel(float* __restrict__ ws) {
  int i = blockIdx.x * blockDim.x + threadIdx.x;
  if (i < ZERO_LEN) ws[ZERO_BEG + i] = 0.0f;
}

// squared norms of all 8192 target rows (first 4096 are gen)
__global__ __launch_bounds__(256) void rownorm_kernel(const float* __restrict__ gen,
                                                      const float* __restrict__ pos,
                                                      float* __restrict__ ws) {
  const int gw   = (blockIdx.x * 256 + threadIdx.x) >> 5;
  const int lane = threadIdx.x & 31;
  if (gw >= GP) return;
  const float* row = (gw < G) ? (gen + (size_t)gw * DD)
                              : (pos + (size_t)(gw - G) * DD);
  float s = 0.0f;
#pragma unroll
  for (int k = lane; k < DD; k += 32) { float x = row[k]; s += x * x; }
  s += __shfl_xor(s, 1, 32);
  s += __shfl_xor(s, 2, 32);
  s += __shfl_xor(s, 4, 32);
  s += __shfl_xor(s, 8, 32);
  s += __shfl_xor(s, 16, 32);
  if (lane == 0) ws[N2_OFF + gw] = s;
}

// Pass 1: per 16x16 tile of K(i,j) accumulate row sums and col sums per temp.
// grid = (32 j-strips of 256 cols, 256 i-tiles), block = 1 wave.
__global__ __launch_bounds__(32) void sums_kernel(const float* __restrict__ gen,
                                                  const float* __restrict__ pos,
                                                  float* __restrict__ ws) {
  const int lane = threadIdx.x;
  const int nl = lane & 15;   // C/D: column N; A: row m
  const int h  = lane >> 4;
  const int ib  = blockIdx.y * 16;
  const int jsb = blockIdx.x * 256;
  const float* n2t = ws + N2_OFF;
  float* rs = ws + RS_OFF;
  float* cs = ws + CS_OFF;

  float n2row[8];
#pragma unroll
  for (int v = 0; v < 8; ++v) n2row[v] = n2t[ib + v + 8 * h];

  float rowacc[3][8];
#pragma unroll
  for (int t = 0; t < 3; ++t)
#pragma unroll
    for (int v = 0; v < 8; ++v) rowacc[t][v] = 0.0f;

  const float* gA = gen + (size_t)(ib + nl) * DD + 2 * h;

  for (int jt = 0; jt < 16; ++jt) {
    const int jb = jsb + jt * 16;
    const float* tbase = (jb < G) ? (gen + (size_t)jb * DD)
                                  : (pos + (size_t)(jb - G) * DD);
    const float* gB = tbase + (size_t)nl * DD + 2 * h;
    v8f c = {0, 0, 0, 0, 0, 0, 0, 0};
    c = wmma_stile(gA, gB, 0, DD, c);

    const float n2col = n2t[jb + nl];
    float dist8[8];
#pragma unroll
    for (int v = 0; v < 8; ++v) {
      const int mg = ib + v + 8 * h;            // global row i
      float d2 = n2row[v] + n2col - 2.0f * c[v];
      float dd = fast_sqrt(fmaxf(d2, 0.0f)) * 0.0625f;   // / sqrt(256)
      if (jb + nl == mg) dd = 1.0e6f;           // diagonal fill (j==i implies j<G)
      dist8[v] = dd;
    }
#pragma unroll
    for (int t = 0; t < 3; ++t) {
      float colp = 0.0f;
#pragma unroll
      for (int v = 0; v < 8; ++v) {
        float kv = __expf(-dist8[v] * INVTAU[t]);
        rowacc[t][v] += kv;
        colp += kv;
      }
      colp += __shfl_xor(colp, 16, 32);         // combine both row halves
      if (lane < 16) atomicAdd(&cs[t * GP + jb + lane], colp);
    }
  }
#pragma unroll
  for (int t = 0; t < 3; ++t)
#pragma unroll
    for (int v = 0; v < 8; ++v) {
      float rv = rowacc[t][v];
      rv += __shfl_xor(rv, 1, 32);
      rv += __shfl_xor(rv, 2, 32);
      rv += __shfl_xor(rv, 4, 32);
      rv += __shfl_xor(rv, 8, 32);
      if (nl == 0) atomicAdd(&rs[t * G + ib + v + 8 * h], rv);
    }
}

// Pass 2: neg_row_sum / pos_row_sum of normalized kernel nk = K*rsqrt(max(rs*cs,1e-12)).
__global__ __launch_bounds__(32) void rowsplit_kernel(const float* __restrict__ gen,
                                                      const float* __restrict__ pos,
                                                      float* __restrict__ ws) {
  const int lane = threadIdx.x;
  const int nl = lane & 15;
  const int h  = lane >> 4;
  const int ib  = blockIdx.y * 16;
  const int jsb = blockIdx.x * 256;
  const float* n2t = ws + N2_OFF;
  const float* rs  = ws + RS_OFF;
  const float* cs  = ws + CS_OFF;
  float* dst = (jsb < G) ? (ws + NRS_OFF) : (ws + PRS_OFF);

  float n2row[8];
  float rsm[3][8];
#pragma unroll
  for (int v = 0; v < 8; ++v) n2row[v] = n2t[ib + v + 8 * h];
#pragma unroll
  for (int t = 0; t < 3; ++t)
#pragma unroll
    for (int v = 0; v < 8; ++v) rsm[t][v] = rs[t * G + ib + v + 8 * h];

  float halfacc[3][8];
#pragma unroll
  for (int t = 0; t < 3; ++t)
#pragma unroll
    for (int v = 0; v < 8; ++v) halfacc[t][v] = 0.0f;

  const float* gA = gen + (size_t)(ib + nl) * DD + 2 * h;

  for (int jt = 0; jt < 16; ++jt) {
    const int jb = jsb + jt * 16;
    const float* tbase = (jb < G) ? (gen + (size_t)jb * DD)
                                  : (pos + (size_t)(jb - G) * DD);
    const float* gB = tbase + (size_t)nl * DD + 2 * h;
    v8f c = {0, 0, 0, 0, 0, 0, 0, 0};
    c = wmma_stile(gA, gB, 0, DD, c);

    const float n2col = n2t[jb + nl];
    float dist8[8];
#pragma unroll
    for (int v = 0; v < 8; ++v) {
      const int mg = ib + v + 8 * h;
      float d2 = n2row[v] + n2col - 2.0f * c[v];
      float dd = fast_sqrt(fmaxf(d2, 0.0f)) * 0.0625f;
      if (jb + nl == mg) dd = 1.0e6f;
      dist8[v] = dd;
    }
#pragma unroll
    for (int t = 0; t < 3; ++t) {
      const float csv = cs[t * GP + jb + nl];
#pragma unroll
      for (int v = 0; v < 8; ++v) {
        float kv = __expf(-dist8[v] * INVTAU[t]);
        halfacc[t][v] += kv * fast_rsqrt(fmaxf(rsm[t][v] * csv, 1e-12f));
      }
    }
  }
#pragma unroll
  for (int t = 0; t < 3; ++t)
#pragma unroll
    for (int v = 0; v < 8; ++v) {
      float rv = halfacc[t][v];
      rv += __shfl_xor(rv, 1, 32);
      rv += __shfl_xor(rv, 2, 32);
      rv += __shfl_xor(rv, 4, 32);
      rv += __shfl_xor(rv, 8, 32);
      if (nl == 0) atomicAdd(&dst[t * G + ib + v + 8 * h], rv);
    }
}

// Pass 3: V_tau[i,:] = sum_j W[i,j] * targets[j,:],
//   W = (j>=G ? +neg_row_sum : -pos_row_sum) * K * rsqrt(max(rs*cs,1e-12)).
// 8 waves share one i-tile. gen i-tile staged once in LDS; each 16-row target
// tile staged per step with coalesced float4 loads (reused by S-partial WMMAs
// and all 24 drift WMMAs); next tile prefetched (global_prefetch).
__global__ __launch_bounds__(256) void drift_kernel(const float* __restrict__ gen,
                                                    const float* __restrict__ pos,
                                                    float* __restrict__ ws) {
  __shared__ float sA[16 * DD];   // 16 KB : gen i-tile rows
  __shared__ float sT[16 * DD];   // 16 KB : target j-tile rows
  __shared__ float sP[8 * 256];   //  8 KB : S partials
  __shared__ float sD[256];       //  1 KB : dist tile
  const int tid  = threadIdx.x;
  const int w    = tid >> 5;
  const int lane = tid & 31;
  const int nl   = lane & 15;
  const int h    = lane >> 4;
  const int ib   = blockIdx.x * 16;
  const int sr   = tid >> 4;          // staging: row 0..15
  const int sc0  = (tid & 15) * 16;   // staging: col base

  const float* n2t = ws + N2_OFF;
  const float* rs  = ws + RS_OFF;
  const float* cs  = ws + CS_OFF;
  const float* nrs = ws + NRS_OFF;
  const float* prs = ws + PRS_OFF;
  float* Vb  = ws + V_OFF;
  float* ssq = ws + SSQ_OFF;

  float rsm[3], nrsm[3], prsm[3];   // row scalars for A-fragment row i = ib+nl
#pragma unroll
  for (int t = 0; t < 3; ++t) {
    rsm[t]  = rs [t * G + ib + nl];
    nrsm[t] = nrs[t * G + ib + nl];
    prsm[t] = prs[t * G + ib + nl];
  }

  int pt[6], pd[6];
#pragma unroll
  for (int q = 0; q < 6; ++q) { int p = w + 8 * q; pt[q] = p >> 4; pd[q] = p & 15; }

  v8f acc[6];
#pragma unroll
  for (int q = 0; q < 6; ++q)
#pragma unroll
    for (int v = 0; v < 8; ++v) acc[q][v] = 0.0f;

  {  // stage gen i-tile once (coalesced 128-bit)
    const float4* src = (const float4*)(gen + (size_t)(ib + sr) * DD + sc0);
    float4* dst = (float4*)(&sA[sr * DD + sc0]);
#pragma unroll
    for (int x = 0; x < 4; ++x) dst[x] = src[x];
  }
  __syncthreads();

  for (int jb = 0; jb < GP; jb += 16) {
    const float* tbase = (jb < G) ? (gen + (size_t)jb * DD)
                                  : (pos + (size_t)(jb - G) * DD);
    {  // stage target j-tile (coalesced 128-bit) + prefetch next tile
      const float4* src = (const float4*)(tbase + (size_t)sr * DD + sc0);
      float4* dst = (float4*)(&sT[sr * DD + sc0]);
#pragma unroll
      for (int x = 0; x < 4; ++x) dst[x] = src[x];
      if (jb + 16 < GP) {
        const int jn = jb + 16;
        const float* nb = (jn < G) ? (gen + (size_t)jn * DD)
                                   : (pos + (size_t)(jn - G) * DD);
        __builtin_prefetch(nb + (size_t)sr * DD + sc0, 0, 1);
      }
    }
    __syncthreads();

    {  // cooperative S tile from LDS: wave w covers K slice [32w, 32w+32)
      v8f cp = {0, 0, 0, 0, 0, 0, 0, 0};
      const int kb = w * 32;
#pragma unroll 8
      for (int k = kb; k < kb + 32; k += 4) {
        v2f a, b;
        a[0] = sA[nl * DD + k + 2 * h];
        a[1] = sA[nl * DD + k + 2 * h + 1];
        b[0] = sT[nl * DD + k + 2 * h];
        b[1] = sT[nl * DD + k + 2 * h + 1];
        cp = __builtin_amdgcn_wmma_f32_16x16x4_f32(false, a, false, b, (short)0,
                                                   cp, false, false);
      }
#pragma unroll
      for (int v = 0; v < 8; ++v) sP[w * 256 + v * 32 + lane] = cp[v];
    }
    __syncthreads();
    {  // reduce partials, convert to dist
      const int v2 = tid >> 5, l2 = tid & 31;
      const int nn = l2 & 15, hh = l2 >> 4;
      const int M = v2 + 8 * hh;
      float s = 0.0f;
#pragma unroll
      for (int w2 = 0; w2 < 8; ++w2) s += sP[w2 * 256 + tid];
      float d2 = n2t[ib + M] + n2t[jb + nn] - 2.0f * s;
      float dd = fast_sqrt(fmaxf(d2, 0.0f)) * 0.0625f;
      if (jb + nn == ib + M) dd = 1.0e6f;
      sD[M * 16 + nn] = dd;
    }
    __syncthreads();

    // build signed normalized-kernel A fragments, all temps, 4 K-steps
    v2f wA[3][4];
#pragma unroll
    for (int ks = 0; ks < 4; ++ks) {
#pragma unroll
      for (int kk = 0; kk < 2; ++kk) {
        const int k  = ks * 4 + 2 * h + kk;
        const int jg = jb + k;
        const float dd = sD[nl * 16 + k];
#pragma unroll
        for (int t = 0; t < 3; ++t) {
          float kv  = __expf(-dd * INVTAU[t]);
          float csv = cs[t * GP + jg];
          float nk  = kv * fast_rsqrt(fmaxf(rsm[t] * csv, 1e-12f));
          float sgn = (jg >= G) ? nrsm[t] : -prsm[t];
          wA[t][ks][kk] = sgn * nk;
        }
      }
    }
#pragma unroll
    for (int q = 0; q < 6; ++q) {
      const int t = pt[q], d = pd[q];
#pragma unroll
      for (int ks = 0; ks < 4; ++ks) {
        const int kr = ks * 4 + 2 * h;
        v2f b;
        b[0] = sT[kr * DD + d * 16 + nl];
        b[1] = sT[(kr + 1) * DD + d * 16 + nl];
        acc[q] = __builtin_amdgcn_wmma_f32_16x16x4_f32(false, wA[t][ks], false, b,
                                                       (short)0, acc[q],
                                                       false, false);
      }
    }
    __syncthreads();
  }

  // store V_tau tiles and accumulate per-temp sum of squares
#pragma unroll
  for (int q = 0; q < 6; ++q) {
    const int t = pt[q], d = pd[q];
    float sq = 0.0f;
#pragma unroll
    for (int v = 0; v < 8; ++v) {
      const int M = v + 8 * h;
      float x = acc[q][v];
      Vb[(size_t)t * ND + (size_t)(ib + M) * DD + d * 16 + nl] = x;
      sq += x * x;
    }
    sq += __shfl_xor(sq, 1, 32);
    sq += __shfl_xor(sq, 2, 32);
    sq += __shfl_xor(sq, 4, 32);
    sq += __shfl_xor(sq, 8, 32);
    sq += __shfl_xor(sq, 16, 32);
    if (lane == 0) atomicAdd(&ssq[t], sq);
  }
}

// V_total = sum_t V_tau / (sqrt(mean(V_tau^2)+1e-8) + 1e-8); accumulate sum(V_total^2)
__global__ __launch_bounds__(256) void total_kernel(float* __restrict__ ws) {
  __shared__ float red[256];
  const int tid = threadIdx.x;
  const int e = blockIdx.x * 256 + tid;
  const float* Vb  = ws + V_OFF;
  const float* ssq = ws + SSQ_OFF;
  float sc[3];
#pragma unroll
  for (int t = 0; t < 3; ++t) {
    float vn = fast_sqrt(ssq[t] * (1.0f / (float)ND) + 1e-8f);
    sc[t] = 1.0f / (vn + 1e-8f);
  }
  float vt = 0.0f;
#pragma unroll
  for (int t = 0; t < 3; ++t) vt += Vb[(size_t)t * ND + e] * sc[t];
  red[tid] = vt * vt;
  __syncthreads();
  for (int s = 128; s > 0; s >>= 1) {
    if (tid < s) red[tid] += red[tid + s];
    __syncthreads();
  }
  if (tid == 0) atomicAdd(ws + ACC_OFF, red[0]);
}

__global__ void finalize_kernel(const float* __restrict__ ws, float* __restrict__ out) {
  out[0] = ws[ACC_OFF] * (1.0f / (float)ND);
}

extern "C" void kernel_launch(void* const* d_in, const int* in_sizes, int n_in,
                              void* d_out, int out_size, void* d_ws, size_t ws_size,
                              hipStream_t stream) {
  (void)in_sizes; (void)n_in; (void)out_size; (void)ws_size;
  const float* gen = (const float*)d_in[0];
  const float* pos = (const float*)d_in[1];
  float* ws  = (float*)d_ws;
  float* out = (float*)d_out;

  zero_kernel<<<(ZERO_LEN + 255) / 256, 256, 0, stream>>>(ws);
  rownorm_kernel<<<GP / 8, 256, 0, stream>>>(gen, pos, ws);
  sums_kernel<<<dim3(32, G / 16), 32, 0, stream>>>(gen, pos, ws);
  rowsplit_kernel<<<dim3(32, G / 16), 32, 0, stream>>>(gen, pos, ws);
  drift_kernel<<<G / 16, 256, 0, stream>>>(gen, pos, ws);
  total_kernel<<<ND / 256, 256, 0, stream>>>(ws);
  finalize_kernel<<<1, 1, 0, stream>>>(ws, out);
}